// GraphRelativeError_40346922778983
// MI455X (gfx1250) — compile-verified
//
#include <hip/hip_runtime.h>
#include <hip/hip_bf16.h>

#define EPSILON 0.1f
#define SCALE   10000.0f
#define G_MAX   1024
#define BLOCK   256

typedef __attribute__((ext_vector_type(2))) float v2f;
typedef __attribute__((ext_vector_type(8))) float v8f;

// Relative error term: |p-t| / (|t|+eps) via v_rcp_f32 (~1 ulp) + v_mul.
// Avoids the ~10-op IEEE divide expansion in the bandwidth-bound hot loop.
__device__ __forceinline__ float rel_term(float p, float t) {
    return fabsf(p - t) * __builtin_amdgcn_rcpf(fabsf(t) + EPSILON);
}

// ---------------------------------------------------------------------------
// Wave-level (wave32) fp32 sum. Uses V_WMMA_F32_16X16X4_F32 as a deterministic
// cross-lane reducer: A[m][0]=acc_m (lanes 0-15), A[m][2]=acc_{m+16}
// (lanes 16-31), other K slots zero; B = all ones. D[m][n] = acc_m + acc_{m+16}
// for every n. Summing the 8 D VGPRs in-lane gives rows 0-7 (lanes 0-15) and
// rows 8-15 (lanes 16-31); combining lane 0 and lane 16 yields the wave total.
// Hardware accumulation order is fixed -> bitwise deterministic.
// ---------------------------------------------------------------------------
__device__ __forceinline__ float wave_sum32(float acc) {
#if __has_builtin(__builtin_amdgcn_wmma_f32_16x16x4_f32)
    v2f a; a.x = acc;  a.y = 0.0f;
    v2f b; b.x = 1.0f; b.y = 1.0f;
    v8f c = {};
    c = __builtin_amdgcn_wmma_f32_16x16x4_f32(
            /*neg_a=*/false, a, /*neg_b=*/false, b,
            /*c_mod=*/(short)0, c, /*reuse_a=*/false, /*reuse_b=*/false);
    float s = ((c[0] + c[1]) + (c[2] + c[3])) + ((c[4] + c[5]) + (c[6] + c[7]));
    return __shfl(s, 0, 32) + __shfl(s, 16, 32);
#else
    for (int off = 16; off > 0; off >>= 1)
        acc += __shfl_down(acc, off, 32);
    return __shfl(acc, 0, 32);
#endif
}

// ---------------------------------------------------------------------------
// Kernel A: segment boundaries via binary search over the sorted batch array.
// seg[g] = first index i with batch[i] >= g, seg[ng] = n.
// ---------------------------------------------------------------------------
__global__ void seg_bounds_kernel(const int* __restrict__ batch, int n,
                                  const int* __restrict__ ngp,
                                  int* __restrict__ seg) {
    const int g  = blockIdx.x * blockDim.x + threadIdx.x;
    const int ng = *ngp;
    if (g > ng) return;
    if (g == 0)  { seg[0]  = 0; return; }
    if (g == ng) { seg[ng] = n; return; }
    int lo = 0, hi = n;
    while (lo < hi) {
        int mid = (lo + hi) >> 1;
        if (batch[mid] < g) lo = mid + 1; else hi = mid;
    }
    seg[g] = lo;
}

// ---------------------------------------------------------------------------
// Kernel B: one workgroup per graph. Stream [start,end) of pred/target with
// aligned float4 loads, accumulate per-thread rel-error partials, reduce with
// WMMA per wave, then a tiny serial combine of the 8 wave sums.
// ---------------------------------------------------------------------------
__global__ void __launch_bounds__(BLOCK)
seg_mean_kernel(const float* __restrict__ pred,
                const float* __restrict__ target,
                const int* __restrict__ seg,
                const int* __restrict__ ngp,
                float* __restrict__ means) {
    const int g  = blockIdx.x;
    const int ng = *ngp;
    if (g >= ng) return;                 // uniform per block

    const int start = seg[g];
    const int end   = seg[g + 1];
    const int tid   = threadIdx.x;

    float acc = 0.0f;

    // Head: [start, a0) — up to 3 unaligned elements.
    int a0 = (start + 3) & ~3;
    if (a0 > end) a0 = end;
    {
        int i = start + tid;
        if (i < a0) acc += rel_term(pred[i], target[i]);
    }

    // Body: 16B-aligned float4 groups (pred/target bases are 16B aligned,
    // a0 % 4 == 0).
    const int nvec = (end - a0) >> 2;
    const int vend = a0 + (nvec << 2);
    const float4* __restrict__ p4 = reinterpret_cast<const float4*>(pred + a0);
    const float4* __restrict__ t4 = reinterpret_cast<const float4*>(target + a0);
    for (int v = tid; v < nvec; v += BLOCK) {
        float4 p = p4[v];
        float4 t = t4[v];
        acc += rel_term(p.x, t.x);
        acc += rel_term(p.y, t.y);
        acc += rel_term(p.z, t.z);
        acc += rel_term(p.w, t.w);
    }

    // Tail: [vend, end) — up to 3 elements.
    {
        int i = vend + tid;
        if (i < end) acc += rel_term(pred[i], target[i]);
    }

    // Wave reduction (EXEC all-ones here: no divergent returns above).
    float wtot = wave_sum32(acc);

    __shared__ float wsum[BLOCK / 32];
    const int wid  = tid >> 5;
    const int lane = tid & 31;
    if (lane == 0) wsum[wid] = wtot;
    __syncthreads();

    if (tid == 0) {
        float t = 0.0f;
        #pragma unroll
        for (int i = 0; i < BLOCK / 32; ++i) t += wsum[i];
        means[g] = t / (float)(end - start);   // once per block: exact divide
    }
}

// ---------------------------------------------------------------------------
// Kernel C: mean of the per-graph means, times SCALE.
// ---------------------------------------------------------------------------
__global__ void __launch_bounds__(BLOCK)
final_mean_kernel(const float* __restrict__ means,
                  const int* __restrict__ ngp,
                  float* __restrict__ out) {
    const int ng = *ngp;
    __shared__ float sm[BLOCK];
    float acc = 0.0f;
    for (int i = threadIdx.x; i < ng; i += BLOCK) acc += means[i];
    sm[threadIdx.x] = acc;
    __syncthreads();
    for (int s = BLOCK / 2; s > 0; s >>= 1) {
        if (threadIdx.x < s) sm[threadIdx.x] += sm[threadIdx.x + s];
        __syncthreads();
    }
    if (threadIdx.x == 0) out[0] = (sm[0] / (float)ng) * SCALE;
}

// ---------------------------------------------------------------------------
// Host launcher. Inputs: pred(f32,N), target(f32,N), batch(i32,N), x(unused),
// num_graphs(i32 scalar). Output: f32 scalar.
// ---------------------------------------------------------------------------
extern "C" void kernel_launch(void* const* d_in, const int* in_sizes, int n_in,
                              void* d_out, int out_size, void* d_ws, size_t ws_size,
                              hipStream_t stream) {
    (void)n_in; (void)out_size; (void)ws_size;
    const float* pred   = (const float*)d_in[0];
    const float* target = (const float*)d_in[1];
    const int*   batch  = (const int*)d_in[2];
    const int*   ngp    = (const int*)d_in[4];
    const int    n      = in_sizes[0];

    int*   seg   = (int*)d_ws;                          // (G_MAX+1) ints
    float* means = (float*)((char*)d_ws + 8192);        // G_MAX floats

    seg_bounds_kernel<<<(G_MAX + 1 + BLOCK - 1) / BLOCK, BLOCK, 0, stream>>>(
        batch, n, ngp, seg);
    seg_mean_kernel<<<G_MAX, BLOCK, 0, stream>>>(pred, target, seg, ngp, means);
    final_mean_kernel<<<1, BLOCK, 0, stream>>>(means, ngp, (float*)d_out);
}